// LgcnConv_49581102465506
// MI455X (gfx1250) — compile-verified
//
#include <hip/hip_runtime.h>
#include <stdint.h>

// ---------------- tunables ----------------
constexpr int D              = 128;           // feature dim (floats)
constexpr int ROWBYTES       = D * 4;         // 512 B per feature row
constexpr int BATCH          = 16;            // edges per async batch
constexpr int BUF_BYTES      = BATCH * ROWBYTES;   // 8 KB per buffer
constexpr int WAVES_PER_BLOCK = 4;
constexpr int THREADS        = WAVES_PER_BLOCK * 32;
constexpr int EDGES_PER_WAVE = 512;

// ---------------- zero-init output ----------------
__global__ void lgcn_zero_kernel(float4* __restrict__ out4, int n4) {
    int i = blockIdx.x * blockDim.x + threadIdx.x;
    if (i < n4) out4[i] = make_float4(0.f, 0.f, 0.f, 0.f);
}

__device__ __forceinline__ float readlane_f(float v, int l) {
    return __builtin_bit_cast(float,
        __builtin_amdgcn_readlane(__builtin_bit_cast(int, v), l));
}

// ---------------- main SpMM kernel ----------------
__global__ __launch_bounds__(THREADS)
void lgcn_spmm_kernel(const int*   __restrict__ rows,
                      const int*   __restrict__ cols,
                      const float* __restrict__ vals,
                      const float* __restrict__ feat,
                      float*       __restrict__ out,
                      int n_edges)
{
    __shared__ __align__(16) float smem[WAVES_PER_BLOCK * 2 * BATCH * D]; // 64 KB

    const int lane  = threadIdx.x & 31;
    const int wave  = threadIdx.x >> 5;
    const int gwave = blockIdx.x * WAVES_PER_BLOCK + wave;
    const int e_begin = gwave * EDGES_PER_WAVE;
    if (e_begin >= n_edges) return;
    const int e_end = (e_begin + EDGES_PER_WAVE < n_edges) ? e_begin + EDGES_PER_WAVE : n_edges;
    const int nb    = (e_end - e_begin + BATCH - 1) / BATCH;

    const uint64_t feat_base = (uint64_t)(uintptr_t)feat;
    // LDS byte offset of this wave's staging area (low 32 bits of generic ptr == LDS offset)
    const unsigned smem_base = (unsigned)(uintptr_t)(void*)smem;
    const unsigned lane_lds  = smem_base + (unsigned)wave * (2u * BUF_BYTES) + (unsigned)lane * 16u;
    const unsigned lane16    = (unsigned)lane * 16u;
    const float4* __restrict__ buf =
        (const float4*)(smem + wave * 2 * BATCH * D);

    float4 acc = make_float4(0.f, 0.f, 0.f, 0.f);
    int cur_row = -1;

    // flush accumulator for cur_row via fp32 atomics (uniform branch, rare)
    auto flush = [&]() {
        float* p = out + (size_t)cur_row * D + lane * 4;
        __hip_atomic_fetch_add(p + 0, acc.x, __ATOMIC_RELAXED, __HIP_MEMORY_SCOPE_AGENT);
        __hip_atomic_fetch_add(p + 1, acc.y, __ATOMIC_RELAXED, __HIP_MEMORY_SCOPE_AGENT);
        __hip_atomic_fetch_add(p + 2, acc.z, __ATOMIC_RELAXED, __HIP_MEMORY_SCOPE_AGENT);
        __hip_atomic_fetch_add(p + 3, acc.w, __ATOMIC_RELAXED, __HIP_MEMORY_SCOPE_AGENT);
        acc = make_float4(0.f, 0.f, 0.f, 0.f);
    };

    // all 32 lanes hold a replicated copy of the batch's 16 edge records
    auto load_meta = [&](int b, int& r, int& c, float& v) {
        int idx  = e_begin + b * BATCH + (lane & 15);
        bool ok  = idx < e_end;
        int safe = ok ? idx : (e_end - 1);
        r = rows[safe];
        c = cols[safe];
        float vv = vals[safe];
        v = ok ? vv : 0.f;           // invalid tail edges contribute 0, same row -> no flush
    };

    // issue 16 async 512B row gathers into buffer `parity`
    auto issue = [&](int c_meta, int parity) {
        asm volatile("s_wait_dscnt 0" ::: "memory"); // buffer reuse safety vs pending ds reads
#pragma unroll
        for (int i = 0; i < BATCH; ++i) {
            int ci = __builtin_amdgcn_readlane(c_meta, i);
            unsigned va = (unsigned)ci * (unsigned)ROWBYTES + lane16;
            unsigned la = lane_lds + (unsigned)(parity * BUF_BYTES + i * ROWBYTES);
            asm volatile("global_load_async_to_lds_b128 %0, %1, %2"
                         :: "v"(la), "v"(va), "s"(feat_base) : "memory");
        }
    };

    // consume buffer `parity`: scalar-broadcast row/val, FMA feature row into acc
    auto consume = [&](int r_meta, float v_meta, int parity) {
#pragma unroll
        for (int i = 0; i < BATCH; ++i) {
            int   ri = __builtin_amdgcn_readlane(r_meta, i);
            float vi = readlane_f(v_meta, i);
            if (ri != cur_row) {               // uniform scalar branch
                if (cur_row >= 0) flush();
                cur_row = ri;
            }
            float4 f = buf[parity * (BATCH * D / 4) + i * (D / 4) + lane];
            acc.x = fmaf(vi, f.x, acc.x);
            acc.y = fmaf(vi, f.y, acc.y);
            acc.z = fmaf(vi, f.z, acc.z);
            acc.w = fmaf(vi, f.w, acc.w);
        }
    };

    // software-pipelined double buffer over async batches
    int r0 = 0, c0 = 0, r1 = 0, c1 = 0;
    float v0 = 0.f, v1 = 0.f;
    load_meta(0, r0, c0, v0);
    issue(c0, 0);

    int b = 0;
    for (;;) {
        if (b + 1 < nb) {
            load_meta(b + 1, r1, c1, v1);
            issue(c1, 1);
            asm volatile("s_wait_asynccnt 16" ::: "memory"); // batch b (older 16) done
        } else {
            asm volatile("s_wait_asynccnt 0" ::: "memory");
        }
        consume(r0, v0, 0);
        if (++b >= nb) break;

        if (b + 1 < nb) {
            load_meta(b + 1, r0, c0, v0);
            issue(c0, 0);
            asm volatile("s_wait_asynccnt 16" ::: "memory");
        } else {
            asm volatile("s_wait_asynccnt 0" ::: "memory");
        }
        consume(r1, v1, 1);
        if (++b >= nb) break;
    }

    if (cur_row >= 0) flush();
}

// ---------------- launcher ----------------
extern "C" void kernel_launch(void* const* d_in, const int* in_sizes, int n_in,
                              void* d_out, int out_size, void* d_ws, size_t ws_size,
                              hipStream_t stream) {
    const int*   rows = (const int*)d_in[0];
    const int*   cols = (const int*)d_in[1];
    const float* vals = (const float*)d_in[2];
    const float* feat = (const float*)d_in[3];
    float*       out  = (float*)d_out;
    const int n_edges = in_sizes[0];

    // zero the output (harness poisons it; we accumulate with atomics)
    int n4 = out_size / 4;
    lgcn_zero_kernel<<<(n4 + 255) / 256, 256, 0, stream>>>((float4*)out, n4);

    int total_waves = (n_edges + EDGES_PER_WAVE - 1) / EDGES_PER_WAVE;
    int blocks      = (total_waves + WAVES_PER_BLOCK - 1) / WAVES_PER_BLOCK;
    lgcn_spmm_kernel<<<blocks, THREADS, 0, stream>>>(rows, cols, vals, feat, out, n_edges);
}